// MinCEMultilabelLoss_31525059953025
// MI455X (gfx1250) — compile-verified
//
#include <hip/hip_runtime.h>
#include <math.h>
#include <stdint.h>

// ---------------------------------------------------------------------------
// MinCE multilabel loss, MI455X (gfx1250).
//
// per-row: loss = (max(x) + log sum exp(x - max)) - max_{ml==1} x
// (double log_softmax collapses). Memory-bound: 655 MB streamed once ->
// ~28us floor at 23.3 TB/s. One wave32 per row; CDNA5 async global->LDS
// streaming with an 8-stage ring buffer (7 KB in flight per wave ~= 57 MB
// chip-wide, above the ~16 MB latency-BW product needed to saturate HBM);
// th:TH_LOAD_NT since the stream exceeds the 192 MB L2. Blocked online
// softmax: 5 v_exp_f32 per 4 elements (chunk max + one rescale).
// ---------------------------------------------------------------------------

#define WAVE 32
#define WPB 8     // waves per block
#define NSTAGE 8  // ring-buffer depth (stages in flight)
constexpr int B_ = 8192;
constexpr int C_ = 10000;
constexpr int CHUNK_F = 128;                 // floats per async stage per array (512B)
constexpr int NCHUNK = C_ / CHUNK_F;         // 78
constexpr int TAIL = C_ - NCHUNK * CHUNK_F;  // 16
constexpr float L2E = 1.4426950408889634f;   // log2(e)
constexpr float LN2 = 0.6931471805599453f;

// Async copy of 512B (32 lanes x 16B): global (saddr + per-lane voffset) -> LDS.
// s_wait_dscnt 0 first: all prior ds_loads from the target buffer have
// completed before the async engine may overwrite it. th:TH_LOAD_NT: stream
// is read once and exceeds L2.
__device__ __forceinline__ void async_copy_b128(unsigned lds_byte_addr,
                                                unsigned gl_byte_off,
                                                const void* sbase) {
  asm volatile("s_wait_dscnt 0x0\n\t"
               "global_load_async_to_lds_b128 %0, %1, %2 th:TH_LOAD_NT"
               :
               : "v"(lds_byte_addr), "v"(gl_byte_off), "s"(sbase)
               : "memory");
}

// ASYNCcnt completes in order: <= 2*(NSTAGE-1) outstanding means the oldest
// stage's two loads (x + mask) have landed in LDS.
__device__ __forceinline__ void wait_async_oldest_done() {
  asm volatile("s_wait_asynccnt 0xe" ::: "memory");  // 14 = 2*(NSTAGE-1)
}

// Blocked online log-sum-exp over a 4-element chunk (base-2 domain):
// one rescale exp + 4 element exps instead of 8 exps.
__device__ __forceinline__ void chunk_update(const float4 xv, const float4 mv,
                                             float& m2, float& s, float& mp) {
  // masked max (natural domain)
  mp = fmaxf(mp, (mv.x > 0.5f) ? xv.x : -INFINITY);
  mp = fmaxf(mp, (mv.y > 0.5f) ? xv.y : -INFINITY);
  mp = fmaxf(mp, (mv.z > 0.5f) ? xv.z : -INFINITY);
  mp = fmaxf(mp, (mv.w > 0.5f) ? xv.w : -INFINITY);

  const float t0 = xv.x * L2E, t1 = xv.y * L2E;
  const float t2 = xv.z * L2E, t3 = xv.w * L2E;
  const float lm = fmaxf(fmaxf(t0, t1), fmaxf(t2, t3));
  const float nm = fmaxf(m2, lm);
  const float e = exp2f(t0 - nm) + exp2f(t1 - nm) +
                  exp2f(t2 - nm) + exp2f(t3 - nm);
  s = fmaf(s, exp2f(m2 - nm), e);  // exp2(-inf)=0 handles the first chunk
  m2 = nm;
}

// Single-element online update (tail only).
__device__ __forceinline__ void online_update(float v, float mlv, float& m2,
                                              float& s, float& mp) {
  mp = fmaxf(mp, (mlv > 0.5f) ? v : -INFINITY);
  const float t = v * L2E;
  const float nm = fmaxf(m2, t);
  s = fmaf(s, exp2f(m2 - nm), exp2f(t - nm));
  m2 = nm;
}

__launch_bounds__(WPB* WAVE) __global__
    void MinCEMultilabelLoss_rows_kernel(const float* __restrict__ x,
                                         const float* __restrict__ ml,
                                         float* __restrict__ row_loss) {
  // Per wave: NSTAGE stages x (512B x + 512B mask) = 8 KB of LDS.
  __shared__ __align__(16) float lds[WPB * NSTAGE * 256];

  const int lane = threadIdx.x & (WAVE - 1);
  const int wid = threadIdx.x >> 5;
  const int row = blockIdx.x * WPB + wid;

  const float* xrow = x + (size_t)row * C_;
  const float* mrow = ml + (size_t)row * C_;

  // LDS layout for this wave: stage s at wbase + s*1024 bytes;
  // x-chunk at +0, mask-chunk at +512; lane slot = lane*16.
  float* wptr = &lds[wid * NSTAGE * 256];
  const unsigned wbase = (unsigned)(uintptr_t)(const void*)wptr;
  const unsigned lane16 = (unsigned)lane * 16u;

  // ---- prologue: fill all NSTAGE ring-buffer slots ---------------------
#pragma unroll
  for (int p = 0; p < NSTAGE; ++p) {
    const unsigned goff = (unsigned)p * 512u + lane16;
    const unsigned sbase = wbase + (unsigned)p * 1024u;
    async_copy_b128(sbase + lane16, goff, xrow);
    async_copy_b128(sbase + 512u + lane16, goff, mrow);
  }

  float m2 = -INFINITY;  // running max in base-2 domain
  float s = 0.0f;        // running sum of 2^(t - m2)
  float mp = -INFINITY;  // masked max (natural domain)

  for (int c = 0; c < NCHUNK; ++c) {
    wait_async_oldest_done();  // stage c's two loads have completed
    const int sb = c & (NSTAGE - 1);
    const float4 xv =
        *reinterpret_cast<const float4*>(wptr + sb * 256 + lane * 4);
    const float4 mv =
        *reinterpret_cast<const float4*>(wptr + sb * 256 + 128 + lane * 4);

    chunk_update(xv, mv, m2, s, mp);

    const int nc = c + NSTAGE;
    if (nc < NCHUNK) {  // refill the slot we just consumed
      const unsigned goff = (unsigned)nc * 512u + lane16;
      const unsigned sbase = wbase + (unsigned)sb * 1024u;
      async_copy_b128(sbase + lane16, goff, xrow);
      async_copy_b128(sbase + 512u + lane16, goff, mrow);
    }
  }

  // ---- tail: 16 remaining floats, one per lane on lanes 0..15 ----------
  if (lane < TAIL) {
    const int j = NCHUNK * CHUNK_F + lane;
    online_update(xrow[j], mrow[j], m2, s, mp);
  }

  // ---- wave32 reduction: merge (m2, s) log-sum-exp state + max mp ------
#pragma unroll
  for (int off = 16; off > 0; off >>= 1) {
    const float om = __shfl_xor(m2, off, WAVE);
    const float os = __shfl_xor(s, off, WAVE);
    const float op = __shfl_xor(mp, off, WAVE);
    const float nm = fmaxf(m2, om);
    s = s * exp2f(m2 - nm) + os * exp2f(om - nm);
    m2 = nm;
    mp = fmaxf(mp, op);
  }

  if (lane == 0) {
    // lse (natural) = ln2 * (m2 + log2(s)); loss = lse - masked_max
    row_loss[row] = LN2 * (m2 + log2f(s)) - mp;
  }
}

// Deterministic single-block mean of the 8192 per-row losses.
__launch_bounds__(256) __global__
    void MinCEMultilabelLoss_mean_kernel(const float* __restrict__ rl,
                                         float* __restrict__ out) {
  __shared__ float sm[256];
  float acc = 0.0f;
  for (int i = threadIdx.x; i < B_; i += 256) acc += rl[i];  // fixed order
  sm[threadIdx.x] = acc;
  __syncthreads();
#pragma unroll
  for (int w = 128; w > 0; w >>= 1) {
    if ((int)threadIdx.x < w) sm[threadIdx.x] += sm[threadIdx.x + w];
    __syncthreads();
  }
  if (threadIdx.x == 0) out[0] = sm[0] * (1.0f / (float)B_);
}

extern "C" void kernel_launch(void* const* d_in, const int* in_sizes, int n_in,
                              void* d_out, int out_size, void* d_ws,
                              size_t ws_size, hipStream_t stream) {
  (void)in_sizes;
  (void)n_in;
  (void)out_size;
  (void)ws_size;
  const float* x = (const float*)d_in[0];   // output      [8192, 10000] f32
  const float* ml = (const float*)d_in[1];  // multilabels [8192, 10000] f32
  float* row_loss = (float*)d_ws;           // 8192 floats of scratch
  float* out = (float*)d_out;               // 1 float

  dim3 grid(B_ / WPB);    // 1024 blocks
  dim3 block(WPB * WAVE); // 256 threads = 8 waves
  MinCEMultilabelLoss_rows_kernel<<<grid, block, 0, stream>>>(x, ml, row_loss);
  MinCEMultilabelLoss_mean_kernel<<<1, 256, 0, stream>>>(row_loss, out);
}